// KGTM_60524679135272
// MI455X (gfx1250) — compile-verified
//
#include <hip/hip_runtime.h>
#include <hip/hip_bf16.h>
#include <math.h>

// Problem constants
#define BATCH 2048
#define NNODE 512
#define HDIM  20
#define BH    (BATCH * HDIM)        // 40960
#define TSTEP 3
#define XELEMS ((size_t)BATCH * NNODE * HDIM)   // 20,971,520

typedef float v2f __attribute__((ext_vector_type(2)));
typedef float v8f __attribute__((ext_vector_type(8)));

#define WMMA_F32(a, b, c) \
    __builtin_amdgcn_wmma_f32_16x16x4_f32(false, (a), false, (b), (short)0, (c), false, false)

// ---------------------------------------------------------------------------
// Kernel 0: AT[k][n] = A[n][k]   (1 MB, trivial)
// ---------------------------------------------------------------------------
__global__ void KGTM_transpose_A(const float* __restrict__ A, float* __restrict__ AT) {
    int idx = blockIdx.x * 256 + threadIdx.x;       // 512*512
    int k = idx >> 9;
    int n = idx & 511;
    AT[idx] = A[n * 512 + k];
}

// ---------------------------------------------------------------------------
// Kernel 1: x_l[n][b][h] = x[b][n][h]; nodes = x_l  (node-major state layout)
// ---------------------------------------------------------------------------
__global__ void KGTM_init_x(const float* __restrict__ x,
                            float* __restrict__ x_l,
                            float* __restrict__ nodes) {
    size_t idx = (size_t)blockIdx.x * 256 + threadIdx.x;   // dest-linear over N*B*H
    int h = (int)(idx % HDIM);
    size_t t = idx / HDIM;
    int b = (int)(t % BATCH);
    int n = (int)(t / BATCH);
    float v = x[((size_t)b * NNODE + n) * HDIM + h];
    x_l[idx]   = v;
    nodes[idx] = v;
}

// ---------------------------------------------------------------------------
// Kernel 2: aggregation GEMMs via fp32 WMMA.
//   agg_in  = A   @ nodes   (A-frag from AT -> coalesced)
//   agg_out = A^T @ nodes   (A-frag from A  -> coalesced)
//   Each wave owns one 16-row tile and TWO 16-col tiles (4 accumulators),
//   amortizing A-fragment loads 2x. B-panel K-chunked in LDS, stride 40
//   (2*40 = 80 = 16 mod 64 -> half-wave bank ranges disjoint).
// ---------------------------------------------------------------------------
__global__ void KGTM_gemm_agg(const float* __restrict__ A,
                              const float* __restrict__ AT,
                              const float* __restrict__ nodes,
                              float* __restrict__ agg_in,
                              float* __restrict__ agg_out) {
    __shared__ float Bs[256 * 40];                 // 40 KB
    const int c0   = blockIdx.x * 32;              // 32-col panel (0..1279)
    const int nblk = blockIdx.y * 128;             // 8 waves x 16 rows
    const int tid  = threadIdx.x;
    const int wave = tid >> 5;                     // wave32
    const int lane = tid & 31;
    const int n0 = nblk + wave * 16;
    const int m  = lane & 15;
    const int kb = (lane >> 4) << 1;               // 0 (lanes 0-15) / 2 (lanes 16-31)

    v8f acc0I = {}, acc0O = {}, acc1I = {}, acc1O = {};

    for (int kc = 0; kc < 512; kc += 256) {
        __syncthreads();                           // previous chunk fully consumed
        // cooperative 256x32 B-panel load, float4 (2048 vec4 / 256 thr = 8 each)
        for (int i = tid; i < 2048; i += 256) {
            int k  = i >> 3;
            int j4 = (i & 7) << 2;
            float4 v = *reinterpret_cast<const float4*>(
                &nodes[(size_t)(kc + k) * BH + c0 + j4]);
            *reinterpret_cast<float4*>(&Bs[k * 40 + j4]) = v;
        }
        __syncthreads();

        for (int kk = 0; kk < 256; kk += 4) {
            const int kr0 = kc + kk + kb;
            const int kr1 = kr0 + 1;
            const int l0  = kk + kb;
            const int l1  = l0 + 1;
            v2f aI, aO, b0, b1;
            aI.x = AT[kr0 * 512 + n0 + m];         // A[n0+m][kr0]
            aI.y = AT[kr1 * 512 + n0 + m];
            aO.x = A [kr0 * 512 + n0 + m];         // A^T fragment
            aO.y = A [kr1 * 512 + n0 + m];
            b0.x = Bs[l0 * 40 + m];
            b0.y = Bs[l1 * 40 + m];
            b1.x = Bs[l0 * 40 + 16 + m];
            b1.y = Bs[l1 * 40 + 16 + m];
            __builtin_prefetch(&AT[(kr0 + 32) * 512 + n0 + m], 0, 1);
            __builtin_prefetch(&A [(kr0 + 32) * 512 + n0 + m], 0, 1);
            acc0I = WMMA_F32(aI, b0, acc0I);
            acc0O = WMMA_F32(aO, b0, acc0O);
            acc1I = WMMA_F32(aI, b1, acc1I);
            acc1O = WMMA_F32(aO, b1, acc1O);
        }
    }

    const int hi = lane >> 4;
#pragma unroll
    for (int r = 0; r < 8; ++r) {
        int row = n0 + r + 8 * hi;                 // C/D layout: M = r + 8*(lane>=16)
        size_t base = (size_t)row * BH + c0 + m;
        agg_in [base]      = acc0I[r];
        agg_out[base]      = acc0O[r];
        agg_in [base + 16] = acc1I[r];
        agg_out[base + 16] = acc1O[r];
    }
}

// ---------------------------------------------------------------------------
// Kernel 3: fused GRU gates + state update + output projection.
//   Block = 16n x 16b tile. Compute phase: b-fast lanes -> coalesced reads of
//   node-major buffers. Output staged in LDS (stride 21 pad) and re-mapped so
//   the store phase writes 1280B contiguous runs into out[t][b][n][h].
// ---------------------------------------------------------------------------
__global__ void KGTM_gru_step(const float* __restrict__ agg_in,
                              const float* __restrict__ agg_out,
                              float* __restrict__ nodes,
                              const float* __restrict__ x_l,
                              const float* __restrict__ W3w, const float* __restrict__ W3u,
                              const float* __restrict__ W4w, const float* __restrict__ W4u,
                              const float* __restrict__ W5w, const float* __restrict__ W5u,
                              const float* __restrict__ W_out, const float* __restrict__ b_out,
                              float* __restrict__ out_t) {
    __shared__ float w[4420];
    __shared__ float so[256 * 21];                 // output staging, padded
    // offsets: W3w 0, W3u 800, W4w 1200, W4u 2000, W5w 2400, W5u 3200, Wout 3600, b 4400
    for (int i = threadIdx.x; i < 800; i += 256) { w[i] = W3w[i]; w[1200 + i] = W4w[i]; }
    for (int i = threadIdx.x; i < 800; i += 256) { w[2400 + i] = W5w[i]; w[3600 + i] = W_out[i]; }
    for (int i = threadIdx.x; i < 400; i += 256) { w[800 + i] = W3u[i]; w[2000 + i] = W4u[i]; w[3200 + i] = W5u[i]; }
    for (int i = threadIdx.x; i < 20; i += 256) { w[4400 + i] = b_out[i]; }
    __syncthreads();

    const int tid = threadIdx.x;
    const int bt  = blockIdx.x & 127;              // 128 b-tiles
    const int nt  = blockIdx.x >> 7;               // 32 n-tiles
    const int b_l = tid & 15;
    const int n_l = tid >> 4;
    const int n = nt * 16 + n_l;
    const int b = bt * 16 + b_l;

    float* nrow = nodes + (size_t)n * BH + b * HDIM;
    const float* ai = agg_in  + (size_t)n * BH + b * HDIM;
    const float* ao = agg_out + (size_t)n * BH + b * HDIM;

    float fn[HDIM], av[2 * HDIM];
#pragma unroll
    for (int k = 0; k < HDIM; ++k) fn[k] = nrow[k];
#pragma unroll
    for (int k = 0; k < HDIM; ++k) av[k] = ai[k];
#pragma unroll
    for (int k = 0; k < HDIM; ++k) av[HDIM + k] = ao[k];

    float zv[HDIM], rv[HDIM];
    for (int j = 0; j < HDIM; ++j) {
        float s = 0.f;
#pragma unroll
        for (int k = 0; k < 2 * HDIM; ++k) s += av[k] * w[j * 40 + k];
#pragma unroll
        for (int k = 0; k < HDIM; ++k) s += fn[k] * w[800 + j * 20 + k];
        zv[j] = 1.f / (1.f + __expf(-s));
    }
    for (int j = 0; j < HDIM; ++j) {
        float s = 0.f;
#pragma unroll
        for (int k = 0; k < 2 * HDIM; ++k) s += av[k] * w[1200 + j * 40 + k];
#pragma unroll
        for (int k = 0; k < HDIM; ++k) s += fn[k] * w[2000 + j * 20 + k];
        rv[j] = 1.f / (1.f + __expf(-s));
    }
    float tv[HDIM];
#pragma unroll
    for (int k = 0; k < HDIM; ++k) tv[k] = rv[k] * fn[k];
    for (int j = 0; j < HDIM; ++j) {
        float s = 0.f;
#pragma unroll
        for (int k = 0; k < 2 * HDIM; ++k) s += av[k] * w[2400 + j * 40 + k];
#pragma unroll
        for (int k = 0; k < HDIM; ++k) s += tv[k] * w[3200 + j * 20 + k];
        float hv = tanhf(s);
        fn[j] = (1.f - zv[j]) * fn[j] + zv[j] * hv;   // hv needs only av/tv -> in-place ok
    }
#pragma unroll
    for (int k = 0; k < HDIM; ++k) nrow[k] = fn[k];

    const float* xr = x_l + (size_t)n * BH + b * HDIM;
    float xv[HDIM];
#pragma unroll
    for (int k = 0; k < HDIM; ++k) xv[k] = xr[k];

    // output projection into LDS staging
    float* srow = &so[(n_l * 16 + b_l) * 21];
    for (int j = 0; j < HDIM; ++j) {
        float s = w[4400 + j];
#pragma unroll
        for (int k = 0; k < HDIM; ++k) s += fn[k] * w[3600 + j * 40 + k];
#pragma unroll
        for (int k = 0; k < HDIM; ++k) s += xv[k] * w[3600 + j * 40 + 20 + k];
        srow[j] = s;
    }
    __syncthreads();

    // store phase: n-fast lanes -> contiguous 1280B runs per b
    const int n_w = tid & 15;
    const int b_w = tid >> 4;
    const float* src = &so[(n_w * 16 + b_w) * 21];
    float* orow = out_t + ((size_t)(bt * 16 + b_w) * NNODE + nt * 16 + n_w) * HDIM;
#pragma unroll
    for (int h = 0; h < HDIM; ++h) orow[h] = src[h];
}

// ---------------------------------------------------------------------------
extern "C" void kernel_launch(void* const* d_in, const int* in_sizes, int n_in,
                              void* d_out, int out_size, void* d_ws, size_t ws_size,
                              hipStream_t stream) {
    const float* x     = (const float*)d_in[0];
    const float* A_in  = (const float*)d_in[1];
    const float* W3w   = (const float*)d_in[2];
    const float* W3u   = (const float*)d_in[3];
    const float* W4w   = (const float*)d_in[4];
    const float* W4u   = (const float*)d_in[5];
    const float* W5w   = (const float*)d_in[6];
    const float* W5u   = (const float*)d_in[7];
    const float* W_out = (const float*)d_in[8];
    const float* b_out = (const float*)d_in[9];
    float* out = (float*)d_out;

    float* ws      = (float*)d_ws;
    float* AT      = ws;                       // 262,144 floats
    float* x_l     = AT + 512 * 512;           // 20,971,520
    float* nodes   = x_l + XELEMS;
    float* agg_in  = nodes + XELEMS;
    float* agg_out = agg_in + XELEMS;

    KGTM_transpose_A<<<(512 * 512) / 256, 256, 0, stream>>>(A_in, AT);
    KGTM_init_x<<<(int)(XELEMS / 256), 256, 0, stream>>>(x, x_l, nodes);

    for (int t = 0; t < TSTEP; ++t) {
        KGTM_gemm_agg<<<dim3(BH / 32, NNODE / 128), 256, 0, stream>>>(
            A_in, AT, nodes, agg_in, agg_out);
        KGTM_gru_step<<<(BATCH * NNODE) / 256, 256, 0, stream>>>(
            agg_in, agg_out, nodes, x_l,
            W3w, W3u, W4w, W4u, W5w, W5u, W_out, b_out,
            out + (size_t)t * XELEMS);
    }
}